// StructuralRegularization_23210003268197
// MI455X (gfx1250) — compile-verified
//
#include <hip/hip_runtime.h>
#include <math.h>

typedef float v2f __attribute__((ext_vector_type(2)));
typedef float v8f __attribute__((ext_vector_type(8)));

#define N_NODES 50000
#define D_FEAT  128
#define K_NEG   5

// t = 1 / (1 + exp(x))  -- computed with v_exp_f32 + v_rcp_f32 (no IEEE divide
// expansion). For the pos loss: (sigmoid(d)-1)^2 == sigmoid(-d)^2 == invsig(d)^2.
// For the neg loss: sigmoid(d)^2 == invsig(-d)^2.
__device__ __forceinline__ float invsig(float x) {
    return __builtin_amdgcn_rcpf(1.0f + __expf(x));
}

// ---- workspace init: zero the accumulator and the has-neighbor flags ----
__global__ void sr_init_kernel(float* acc, int* flags, int nflags) {
    int i = blockIdx.x * blockDim.x + threadIdx.x;
    if (i == 0) *acc = 0.0f;
    if (i < nflags) flags[i] = 0;
}

// ---- mark nodes that have at least one out-edge (racy stores of 1: fine) ----
__global__ void sr_flag_kernel(const int* __restrict__ src, int nEdges, int* flags) {
    int e = blockIdx.x * blockDim.x + threadIdx.x;
    if (e < nEdges) flags[src[e]] = 1;
}

// ---- positive loss: sum over edges of (sigmoid(<e_s,e_d>) - 1)^2 ----
// One wave handles batches of 16 edges. A = 16 src rows, B^T = 16 dst rows.
// 32x v_wmma_f32_16x16x4_f32 accumulate the 16x16 gram; we keep the diagonal.
__global__ void sr_pos_kernel(const float* __restrict__ emb,
                              const int* __restrict__ ei,
                              int nEdges, float* acc) {
    const int lane   = threadIdx.x & 31;
    const int wave   = (blockIdx.x * blockDim.x + threadIdx.x) >> 5;
    const int nWaves = (gridDim.x * blockDim.x) >> 5;
    const int nBatches = (nEdges + 15) >> 4;

    const int m     = lane & 15;   // row of A / col of B this lane feeds
    const int khalf = lane >> 4;   // which K-pair this lane holds

    float local = 0.0f;
    for (int b = wave; b < nBatches; b += nWaves) {
        const int base = b << 4;
        int em = base + m;
        if (em >= nEdges) em = nEdges - 1;
        const int srow = ei[em] * D_FEAT;           // src row base
        const int drow = ei[nEdges + em] * D_FEAT;  // dst row base

        v8f c = {};
        #pragma unroll 4
        for (int kc = 0; kc < D_FEAT / 4; ++kc) {
            const int off = kc * 4 + khalf * 2;
            v2f av = *(const v2f*)(emb + srow + off);
            v2f bv = *(const v2f*)(emb + drow + off);
            c = __builtin_amdgcn_wmma_f32_16x16x4_f32(false, av, false, bv,
                                                      (short)0, c, false, false);
        }

        // diagonal of the 16x16 f32 C tile (wave32 layout)
        int dsel = (lane < 8) ? lane : ((lane >= 24) ? (lane - 24) : -1);
        float dot = 0.0f;
        #pragma unroll
        for (int v = 0; v < 8; ++v) dot = (dsel == v) ? c[v] : dot;
        int dIdx = (lane < 8) ? lane : ((lane >= 24) ? (lane - 16) : -1);
        if (dIdx >= 0) {
            int e = base + dIdx;
            if (e < nEdges) {
                float t = invsig(dot);  // (sigmoid(dot)-1)^2 == t^2
                local += t * t;
            }
        }
    }
    #pragma unroll
    for (int off = 16; off > 0; off >>= 1)
        local += __shfl_down(local, off, 32);
    if (lane == 0) atomicAdd(acc, local);
}

// ---- negative loss: sum over (i,j) of has_nb[i] * sigmoid(<e_i,e_neg[i,j]>)^2 ----
__global__ void sr_neg_kernel(const float* __restrict__ emb,
                              const int* __restrict__ negs,
                              const int* __restrict__ flags,
                              int nPairs, float* acc) {
    const int lane   = threadIdx.x & 31;
    const int wave   = (blockIdx.x * blockDim.x + threadIdx.x) >> 5;
    const int nWaves = (gridDim.x * blockDim.x) >> 5;
    const int nBatches = (nPairs + 15) >> 4;

    const int m     = lane & 15;
    const int khalf = lane >> 4;

    float local = 0.0f;
    for (int b = wave; b < nBatches; b += nWaves) {
        const int base = b << 4;
        int pm = base + m;
        if (pm >= nPairs) pm = nPairs - 1;
        const int srow = (pm / K_NEG) * D_FEAT;  // node i row
        const int drow = negs[pm] * D_FEAT;      // sampled row

        v8f c = {};
        #pragma unroll 4
        for (int kc = 0; kc < D_FEAT / 4; ++kc) {
            const int off = kc * 4 + khalf * 2;
            v2f av = *(const v2f*)(emb + srow + off);
            v2f bv = *(const v2f*)(emb + drow + off);
            c = __builtin_amdgcn_wmma_f32_16x16x4_f32(false, av, false, bv,
                                                      (short)0, c, false, false);
        }

        int dsel = (lane < 8) ? lane : ((lane >= 24) ? (lane - 24) : -1);
        float dot = 0.0f;
        #pragma unroll
        for (int v = 0; v < 8; ++v) dot = (dsel == v) ? c[v] : dot;
        int dIdx = (lane < 8) ? lane : ((lane >= 24) ? (lane - 16) : -1);
        if (dIdx >= 0) {
            int p = base + dIdx;
            if (p < nPairs && flags[p / K_NEG]) {
                float t = invsig(-dot);  // sigmoid(dot)^2 == t^2
                local += t * t;
            }
        }
    }
    #pragma unroll
    for (int off = 16; off > 0; off >>= 1)
        local += __shfl_down(local, off, 32);
    if (lane == 0) atomicAdd(acc, local);
}

__global__ void sr_final_kernel(const float* acc, float* out, float invn) {
    if (blockIdx.x == 0 && threadIdx.x == 0) out[0] = acc[0] * invn;
}

extern "C" void kernel_launch(void* const* d_in, const int* in_sizes, int n_in,
                              void* d_out, int out_size, void* d_ws, size_t ws_size,
                              hipStream_t stream) {
    const float* emb  = (const float*)d_in[0];  // [N_NODES, D_FEAT] f32
    const int*   ei   = (const int*)d_in[1];    // [2, N_EDGES] int
    const int*   negs = (const int*)d_in[2];    // [N_NODES, K_NEG] int

    const int nEdges = in_sizes[1] / 2;
    const int nNodes = in_sizes[2] / K_NEG;
    const int nPairs = in_sizes[2];

    float* acc   = (float*)d_ws;
    int*   flags = (int*)((char*)d_ws + 256);
    float* out   = (float*)d_out;

    // 1) zero accumulator + flags (must happen every call: graph replays)
    {
        int blocks = (nNodes + 255) / 256;
        sr_init_kernel<<<blocks, 256, 0, stream>>>(acc, flags, nNodes);
    }
    // 2) mark nodes with out-edges
    {
        int blocks = (nEdges + 255) / 256;
        sr_flag_kernel<<<blocks, 256, 0, stream>>>(ei, nEdges, flags);
    }
    // 3) positive-edge loss via fp32 WMMA gram diagonals
    sr_pos_kernel<<<256, 256, 0, stream>>>(emb, ei, nEdges, acc);
    // 4) negative-sample loss via fp32 WMMA gram diagonals
    sr_neg_kernel<<<128, 256, 0, stream>>>(emb, negs, flags, nPairs, acc);
    // 5) scale by 1/N
    sr_final_kernel<<<1, 32, 0, stream>>>(acc, out, 1.0f / (float)nNodes);
}